// GraphEGIN_14233521619241
// MI455X (gfx1250) — compile-verified
//
#include <hip/hip_runtime.h>

typedef __attribute__((ext_vector_type(2))) float v2f;
typedef __attribute__((ext_vector_type(8))) float v8f;

#define NNODES   50000
#define NDEG     16
#define FEDGE    16
#define DH       128
#define DCAT     144
#define NGRAPH   64
#define DOUT     10
#define NLAYERS  3
#define BN_SCALE 0.9999950000374997f

// ---------------------------------------------------------------------------
// edge_rep[i][f] = 1 + sum_{k<16} edge_attr[i*16+k][f]
// ---------------------------------------------------------------------------
__global__ void edge_rep_kernel(const float* __restrict__ ea,
                                float* __restrict__ er) {
    int idx = blockIdx.x * blockDim.x + threadIdx.x;
    if (idx >= NNODES * FEDGE) return;
    int i = idx >> 4, f = idx & 15;
    const float* base = ea + (size_t)i * (NDEG * FEDGE) + f;
    float s = 1.0f;
#pragma unroll
    for (int k = 0; k < NDEG; ++k) s += base[k * FEDGE];
    er[idx] = s;
}

// ---------------------------------------------------------------------------
// pooled[i][0:128]   = h[i]*(1+eps) + sum_k h[edge_dst[i*16+k]]
// pooled[i][128:144] = edge_rep[i] + eps
// One wave per node, each lane owns 4 channels (float4).
// ---------------------------------------------------------------------------
__global__ void build_pooled(const float* __restrict__ hin,
                             const float* __restrict__ erep,
                             const int* __restrict__ edge_dst,
                             const float* __restrict__ eps_arr, int layer,
                             float* __restrict__ pooled) {
    int i = blockIdx.x * (blockDim.x >> 5) + (threadIdx.x >> 5);
    int lane = threadIdx.x & 31;
    if (i >= NNODES) return;
    float eps = eps_arr[layer];
    float s = 1.0f + eps;
    float4 a = ((const float4*)(hin + (size_t)i * DH))[lane];
    float4 acc = make_float4(a.x * s, a.y * s, a.z * s, a.w * s);
    const int* dst = edge_dst + (size_t)i * NDEG;
#pragma unroll 4
    for (int k = 0; k < NDEG; ++k) {
        int j = dst[k];
        float4 nb = ((const float4*)(hin + (size_t)j * DH))[lane];
        acc.x += nb.x; acc.y += nb.y; acc.z += nb.z; acc.w += nb.w;
    }
    float* prow = pooled + (size_t)i * DCAT;           // 576B rows: 16B aligned
    ((float4*)prow)[lane] = acc;
    if (lane < FEDGE) prow[DH + lane] = erep[(size_t)i * FEDGE + lane] + eps;
}

// ---------------------------------------------------------------------------
// Out = relu( gamma * ((A@W + bias) * BN_SCALE) + beta )
// A: nrows x K (row-major, K in {144,128}), W: K x 128, Out: nrows x 128.
// Block = 8 waves; block covers 16 rows x 128 cols; wave w owns cols
// [16w,16w+16). A-tile staged in LDS; fp32 WMMA 16x16x4, K/4 steps.
// Fragment layouts per CDNA5 ISA 7.12.2:
//   A 16x4 : lane<16 -> M=lane, v0=K0,v1=K1 ; lane>=16 -> v0=K2,v1=K3
//   B 4x16 : lane<16 -> N=lane, v0=K0,v1=K1 ; lane>=16 -> v0=K2,v1=K3
//   C/D    : vr = row (8*half + r), col = lane&15
// ---------------------------------------------------------------------------
__global__ void gemm_bn_relu(const float* __restrict__ A,
                             const float* __restrict__ W,
                             const float* __restrict__ bias,
                             const float* __restrict__ gamma,
                             const float* __restrict__ beta,
                             float* __restrict__ Out, int nrows, int K) {
    __shared__ float As[16 * DCAT];            // max K = 144
    const int tid  = threadIdx.x;
    const int row0 = blockIdx.x * 16;

    // Stage 16 x K A-tile into LDS (zero-pad rows beyond nrows)
    for (int idx = tid; idx < 16 * K; idx += 256) {
        int r = idx / K, c = idx - r * K;
        int row = row0 + r;
        As[idx] = (row < nrows) ? A[(size_t)row * K + c] : 0.0f;
    }
    __syncthreads();

    const int lane = tid & 31;
    const int wave = tid >> 5;                 // 0..7 -> column tile
    const int half = lane >> 4;                // 0: K+0/K+1 ; 1: K+2/K+3
    const int l16  = lane & 15;
    const int col  = wave * 16 + l16;

    v8f acc = {};
    for (int kk = 0; kk < K; kk += 4) {
        const int ka = kk + 2 * half;
        v2f a, b;
        a.x = As[l16 * K + ka];
        a.y = As[l16 * K + ka + 1];
        b.x = W[(size_t)ka * DH + col];
        b.y = W[(size_t)(ka + 1) * DH + col];
        acc = __builtin_amdgcn_wmma_f32_16x16x4_f32(
            /*neg_a=*/false, a, /*neg_b=*/false, b,
            /*c_mod=*/(short)0, acc, /*reuse_a=*/false, /*reuse_b=*/false);
    }

    const float g  = gamma[col];
    const float bt = beta[col];
    const float bs = bias[col];
#pragma unroll
    for (int r = 0; r < 8; ++r) {
        int row = row0 + 8 * half + r;
        if (row < nrows) {
            float v = g * ((acc[r] + bs) * BN_SCALE) + bt;
            Out[(size_t)row * DH + col] = fmaxf(v, 0.0f);
        }
    }
}

// ---------------------------------------------------------------------------
// Deterministic per-graph pooling: batch is sorted, so block g binary-searches
// its node range and reduces sequentially (no float atomics).
// ---------------------------------------------------------------------------
__global__ void pool_rep(const float* __restrict__ h,
                         const int* __restrict__ batch,
                         float* __restrict__ out) {
    const int g = blockIdx.x;
    const int c = threadIdx.x;                 // 128 channels
    int lo = 0, hi = NNODES;
    while (lo < hi) { int m = (lo + hi) >> 1; if (batch[m] < g) lo = m + 1; else hi = m; }
    const int start = lo;
    hi = NNODES;
    while (lo < hi) { int m = (lo + hi) >> 1; if (batch[m] < g + 1) lo = m + 1; else hi = m; }
    const int end = lo;
    float acc = 0.0f;
    for (int i = start; i < end; ++i) acc += h[(size_t)i * DH + c];
    out[(size_t)g * DH + c] = acc;
}

// ---------------------------------------------------------------------------
// score[g][o] = sum_l ( pooled_reps[l][g] @ Wp[l] + bp[l] )
// ---------------------------------------------------------------------------
__global__ void readout_kernel(const float* __restrict__ pr,
                               const float* __restrict__ Wp,
                               const float* __restrict__ bp,
                               float* __restrict__ score) {
    int idx = blockIdx.x * blockDim.x + threadIdx.x;
    if (idx >= NGRAPH * DOUT) return;
    int g = idx / DOUT, o = idx - g * DOUT;
    float acc = 0.0f;
    for (int l = 0; l < NLAYERS + 1; ++l) {
        acc += bp[l * DOUT + o];
        const float* p = pr + ((size_t)l * NGRAPH + g) * DH;
        const float* w = Wp + (size_t)l * DH * DOUT + o;
        for (int c = 0; c < DH; ++c) acc += p[c] * w[c * DOUT];
    }
    score[idx] = acc;
}

// ---------------------------------------------------------------------------
extern "C" void kernel_launch(void* const* d_in, const int* in_sizes, int n_in,
                              void* d_out, int out_size, void* d_ws, size_t ws_size,
                              hipStream_t stream) {
    (void)in_sizes; (void)n_in; (void)out_size; (void)ws_size;
    const float* x         = (const float*)d_in[0];
    const float* edge_attr = (const float*)d_in[1];
    const float* eps       = (const float*)d_in[2];
    const float* W1        = (const float*)d_in[3];
    const float* b1        = (const float*)d_in[4];
    const float* g1        = (const float*)d_in[5];
    const float* be1       = (const float*)d_in[6];
    const float* W2        = (const float*)d_in[7];
    const float* b2        = (const float*)d_in[8];
    const float* bn_g      = (const float*)d_in[9];
    const float* bn_b      = (const float*)d_in[10];
    const float* Wp        = (const float*)d_in[11];
    const float* bp        = (const float*)d_in[12];
    const int*   edge_dst  = (const int*)d_in[14];
    const int*   batch     = (const int*)d_in[15];
    float* out = (float*)d_out;

    // Workspace carve (all regions fully written before any read)
    float* ws     = (float*)d_ws;
    float* er     = ws;                                   // N*16
    float* h      = er     + (size_t)NNODES * FEDGE;      // N*128
    float* pooled = h      + (size_t)NNODES * DH;         // N*144
    float* h1     = pooled + (size_t)NNODES * DCAT;       // N*128
    float* preps  = h1     + (size_t)NNODES * DH;         // 4*64*128

    edge_rep_kernel<<<(NNODES * FEDGE + 255) / 256, 256, 0, stream>>>(edge_attr, er);
    pool_rep<<<NGRAPH, DH, 0, stream>>>(x, batch, preps);

    const float* hin = x;
    for (int l = 0; l < NLAYERS; ++l) {
        build_pooled<<<(NNODES + 7) / 8, 256, 0, stream>>>(hin, er, edge_dst, eps, l, pooled);
        gemm_bn_relu<<<(NNODES + 15) / 16, 256, 0, stream>>>(
            pooled, W1 + (size_t)l * DCAT * DH, b1 + l * DH, g1 + l * DH,
            be1 + l * DH, h1, NNODES, DCAT);
        gemm_bn_relu<<<(NNODES + 15) / 16, 256, 0, stream>>>(
            h1, W2 + (size_t)l * DH * DH, b2 + l * DH, bn_g + l * DH,
            bn_b + l * DH, h, NNODES, DH);
        pool_rep<<<NGRAPH, DH, 0, stream>>>(h, batch, preps + (size_t)(l + 1) * NGRAPH * DH);
        hin = h;
    }
    readout_kernel<<<(NGRAPH * DOUT + 255) / 256, 256, 0, stream>>>(preps, Wp, bp, out);
}